// FundamentalEmbeddingsEncoder_87290915324498
// MI455X (gfx1250) — compile-verified
//
#include <hip/hip_runtime.h>
#include <hip/hip_bf16.h>

// Problem constants (from the reference)
#define N_FUND      8192
#define N_SPLITS    127
#define N_SAMPLES   32
#define DIM         32          // embedding dim
#define IDX_SCALE   64.0f       // OFFSET = 8192 / 127 ~ 64
#define N_POS       (2048 * 100)
#define WAVES_PER_BLOCK 8
#define POS_PER_WAVE    2       // software pipeline depth on ASYNCcnt
#define ROW_BYTES   (DIM * 4)   // 128 B per table row

// 32-bit Murmur3-style finalizer: the sample counter (< 2^23) fits in 32 bits,
// so we avoid splitmix64's three 64-bit multiplies (costly on 32-bit VALU).
__device__ __forceinline__ unsigned hash32(unsigned x) {
    x ^= x >> 16; x *= 0x7FEB352Du;
    x ^= x >> 15; x *= 0x846CA68Bu;
    x ^= x >> 16;
    return x;
}

// Deterministic chi^2(dof) draw -> table row index in [0, 8191].
// counter hash -> Box-Muller normal -> Wilson-Hilferty chi^2. Branch-free.
__device__ __forceinline__ int sample_row(unsigned p, unsigned lane, int dof) {
    unsigned ctr = (p << 5) | lane;
    unsigned h1  = hash32(ctr * 0x9E3779B9u + 0x42D3C0FFu);
    unsigned h2  = hash32(h1 + 0x9E3779B9u);
    float u1 = (float)((h1 >> 8) + 1u) * (1.0f / 16777216.0f);   // (0,1]
    float u2 = (float)(h2 >> 8) * (1.0f / 16777216.0f);          // [0,1)
    float r  = __fsqrt_rn(-2.0f * __logf(u1));
    float z  = r * __cosf(6.28318530718f * u2);
    float fd   = (float)dof;
    float c    = 2.0f / (9.0f * fd);
    float t    = fmaxf(1.0f - c + z * __fsqrt_rn(c), 0.0f);
    float chi2 = fd * t * t * t;                                 // >= 0
    return (int)fminf(chi2 * IDX_SCALE, (float)(N_FUND - 1));
}

// One async B128 moves 16 B/lane; the offset: immediate applies to BOTH the LDS
// destination and the global source (ISA 10.7 async pseudocode), so the whole
// 128 B/lane row gather shares a single VGPR address pair.
#define ASYNC_ROW16(LDS, GADDR, OFF)                                        \
    asm volatile("global_load_async_to_lds_b128 %0, %1, off offset:" #OFF   \
                 :: "v"(LDS), "v"(GADDR) : "memory")

#define ASYNC_ROW128(LDS, GADDR)      \
    ASYNC_ROW16(LDS, GADDR, 0);       \
    ASYNC_ROW16(LDS, GADDR, 16);      \
    ASYNC_ROW16(LDS, GADDR, 32);      \
    ASYNC_ROW16(LDS, GADDR, 48);      \
    ASYNC_ROW16(LDS, GADDR, 64);      \
    ASYNC_ROW16(LDS, GADDR, 80);      \
    ASYNC_ROW16(LDS, GADDR, 96);      \
    ASYNC_ROW16(LDS, GADDR, 112)

__device__ __forceinline__ float max_over_rows(const float* w, unsigned lane) {
    // Iteration s reads 32 consecutive dwords across the wave: bank-conflict-free.
    float acc = w[lane];
    #pragma unroll 8
    for (int s = 1; s < N_SAMPLES; ++s)
        acc = fmaxf(acc, w[(unsigned)s * DIM + lane]);
    return acc;
}

__global__ __launch_bounds__(256) void FundamentalEmbeddingsEncoder_87290915324498_kernel(
    const int*   __restrict__ src32,   // int64 src viewed as int pairs (low word holds value)
    const float* __restrict__ fund,    // [8192, 32] f32
    float*       __restrict__ out)     // [N_POS, 32] f32
{
    // Per-wave staging: 2 positions x 32 rows x 128 B = 8 KB/wave, 64 KB/block
    // (of the 320 KB WGP pool).
    __shared__ __align__(16) float smem[WAVES_PER_BLOCK * POS_PER_WAVE * N_SAMPLES * DIM];

    const unsigned wave = threadIdx.x >> 5;   // wave32 (gfx1250)
    const unsigned lane = threadIdx.x & 31;
    const unsigned p0   = (blockIdx.x * WAVES_PER_BLOCK + wave) * POS_PER_WAVE; // exact cover
    const unsigned p1   = p0 + 1;

    // gfx1250 prefetch path (global_prefetch_b8) for the wave-uniform src values.
    __builtin_prefetch(src32 + 2 * p0, 0, 0);

    int dof0 = src32[2 * p0];            // int64 value fits in low 32 bits (0..126)
    int dof1 = src32[2 * p1];
    dof0 = min(max(dof0, 1), N_SPLITS);
    dof1 = min(max(dof1, 1), N_SPLITS);

    // Lane = sample index s for both pipelined positions.
    const unsigned idx0 = (unsigned)sample_row(p0, lane, dof0);
    const unsigned idx1 = (unsigned)sample_row(p1, lane, dof1);

    unsigned long long fbase = (unsigned long long)(const void*)fund;
    unsigned long long g0 = fbase + (unsigned long long)(idx0 * (unsigned)ROW_BYTES);
    unsigned long long g1 = fbase + (unsigned long long)(idx1 * (unsigned)ROW_BYTES);

    // Generic->LDS: low 32 bits of the flat address are the DS byte address.
    float* slot0 = &smem[((wave * POS_PER_WAVE + 0) * N_SAMPLES + lane) * DIM];
    float* slot1 = &smem[((wave * POS_PER_WAVE + 1) * N_SAMPLES + lane) * DIM];
    unsigned ldsRow0 = (unsigned)(unsigned long long)(const void*)slot0;
    unsigned ldsRow1 = (unsigned)(unsigned long long)(const void*)slot1;

    // Issue both gathers back-to-back: 16 async B128 ops in flight (ASYNCcnt=16).
    ASYNC_ROW128(ldsRow0, g0);
    ASYNC_ROW128(ldsRow1, g1);

    // Async loads complete in order: ASYNCcnt <= 8 means position 0's 4 KB landed.
    asm volatile("s_wait_asynccnt 8" ::: "memory");
    const float* w0 = &smem[(wave * POS_PER_WAVE + 0) * N_SAMPLES * DIM];
    out[p0 * DIM + lane] = max_over_rows(w0, lane);     // overlaps position 1's DMA

    asm volatile("s_wait_asynccnt 0" ::: "memory");
    const float* w1 = &smem[(wave * POS_PER_WAVE + 1) * N_SAMPLES * DIM];
    out[p1 * DIM + lane] = max_over_rows(w1, lane);     // coalesced 128 B stores
}

extern "C" void kernel_launch(void* const* d_in, const int* in_sizes, int n_in,
                              void* d_out, int out_size, void* d_ws, size_t ws_size,
                              hipStream_t stream) {
    (void)in_sizes; (void)n_in; (void)out_size; (void)d_ws; (void)ws_size;
    const int*   src32 = (const int*)d_in[0];    // int64 [2048,100] -> int pairs
    const float* fund  = (const float*)d_in[1];  // f32 [8192,32]
    float*       out   = (float*)d_out;          // f32 [2048,100,32]

    dim3 block(256);                                           // 8 wave32 waves
    dim3 grid(N_POS / (WAVES_PER_BLOCK * POS_PER_WAVE));       // 12,800 blocks... exact cover
    FundamentalEmbeddingsEncoder_87290915324498_kernel<<<grid, block, 0, stream>>>(src32, fund, out);
}